// ChebconvRez_53403623358893
// MI455X (gfx1250) — compile-verified
//
#include <hip/hip_runtime.h>
#include <math.h>

// ---------------------------------------------------------------------------
// ChebConv (K=3) x2 + residual, tuned for MI455X (gfx1250, wave32).
//  - scatter/gather prop passes: float4 gathers, f32 L2 atomics (L2-resident:
//    whole working set ~60MB << 192MB L2)
//  - dense combine sum_k Txk @ Wk done with V_WMMA_F32_16X16X4_F32, one
//    16-node tile per wave, K=96 (3 terms x 32 feats) -> 48 WMMA per tile.
//  - Tx2 = 2*prop(Tx1) - Tx0 folded into the WMMA A-fragment (no extra pass).
// ---------------------------------------------------------------------------

typedef __attribute__((ext_vector_type(2))) float v2f;
typedef __attribute__((ext_vector_type(8))) float v8f;

#define NN 100000
#define NE 1600000
#define DF 32

// ---- degree: deg[row] += (row != col) --------------------------------------
__global__ void deg_kernel(const int* __restrict__ row, const int* __restrict__ col,
                           float* __restrict__ deg, int e_cnt) {
  int e = blockIdx.x * blockDim.x + threadIdx.x;
  if (e >= e_cnt) return;
  int r = row[e], c = col[e];
  if (r != c) atomicAdd(&deg[r], 1.0f);
}

// ---- dinv = deg > 0 ? rsqrt(deg) : 0 ---------------------------------------
__global__ void dinv_kernel(const float* __restrict__ deg, float* __restrict__ dinv, int n) {
  int i = blockIdx.x * blockDim.x + threadIdx.x;
  if (i >= n) return;
  float d = deg[i];
  dinv[i] = (d > 0.0f) ? rsqrtf(d) : 0.0f;
}

// ---- edge weights: w = -(row!=col) * dinv[row]*dinv[col] -------------------
__global__ void w_kernel(const int* __restrict__ row, const int* __restrict__ col,
                         const float* __restrict__ dinv, float* __restrict__ w, int e_cnt) {
  int e = blockIdx.x * blockDim.x + threadIdx.x;
  if (e >= e_cnt) return;
  int r = row[e], c = col[e];
  w[e] = (r != c) ? (-dinv[r] * dinv[c]) : 0.0f;
}

// ---- prop: Y[row] += w * X[col]  (8 lanes per edge, float4 each) -----------
__global__ void prop_kernel(const float* __restrict__ X,
                            const int* __restrict__ row, const int* __restrict__ col,
                            const float* __restrict__ w, float* __restrict__ Y, int e_cnt) {
  int t = blockIdx.x * blockDim.x + threadIdx.x;
  int e = t >> 3;
  if (e >= e_cnt) return;
  int q = (t & 7) * 4;
  float wv = w[e];
  if (wv == 0.0f) return;                 // self-loops contribute nothing
  int c = col[e], r = row[e];
  float4 v = *(const float4*)(X + (size_t)c * DF + q);
  float* dst = Y + (size_t)r * DF + q;
  atomicAdd(dst + 0, wv * v.x);
  atomicAdd(dst + 1, wv * v.y);
  atomicAdd(dst + 2, wv * v.z);
  atomicAdd(dst + 3, wv * v.w);
}

// ---- WMMA combine:  out = relu?( [Tx0|Tx1|2*P1-Tx0] @ [W0;W1;W2] + b (+resid) )
// One wave32 computes one 16-node tile (16x32 output) with f32 WMMA 16x16x4.
__global__ void cheb_wmma_kernel(const float* __restrict__ X0,   // Tx0  (N x 32)
                                 const float* __restrict__ T1,   // Tx1  (N x 32)
                                 const float* __restrict__ P1,   // prop(Tx1)
                                 const float* __restrict__ W,    // 3 x 32 x 32
                                 const float* __restrict__ bias, // 32
                                 const float* __restrict__ resid,// N x 32 or null
                                 float* __restrict__ out,        // N x 32
                                 int n_nodes, int do_relu) {
  const int lane = threadIdx.x & 31;
  const int wave = threadIdx.x >> 5;
  const int tile = blockIdx.x * (blockDim.x >> 5) + wave;   // 16-node tile
  if (tile * 16 >= n_nodes) return;                          // wave-uniform

  const int m    = lane & 15;          // row inside tile (A), or N index (B/C)
  const int half = lane >> 4;          // 0: K={0,1}, 1: K={2,3} within k-step
  const int koff = half * 2;

  // Preload B fragments: B[term][k4][colgrp]; lane holds W[kb+{0,1}][n]
  v2f B[3][8][2];
#pragma unroll
  for (int t = 0; t < 3; ++t)
#pragma unroll
    for (int k4 = 0; k4 < 8; ++k4) {
      const int kb = k4 * 4 + koff;
      const float* wrow = W + t * (32 * 32) + kb * 32;
#pragma unroll
      for (int g = 0; g < 2; ++g) {
        v2f b;
        b.x = wrow[g * 16 + m];
        b.y = wrow[32 + g * 16 + m];
        B[t][k4][g] = b;
      }
    }

  v8f c_lo = {};   // output cols 0..15
  v8f c_hi = {};   // output cols 16..31

  const size_t base = (size_t)(tile * 16 + m) * DF;
  const float* x0r = X0 + base;
  const float* t1r = T1 + base;
  const float* p1r = P1 + base;

#pragma unroll
  for (int k4 = 0; k4 < 8; ++k4) {
    const int kb = k4 * 4 + koff;
    v2f a0, a1, p;
    a0.x = x0r[kb]; a0.y = x0r[kb + 1];
    a1.x = t1r[kb]; a1.y = t1r[kb + 1];
    p.x  = p1r[kb]; p.y  = p1r[kb + 1];
    v2f a2 = 2.0f * p - a0;            // Tx2 = 2*L_hat*Tx1 - Tx0, folded here

    c_lo = __builtin_amdgcn_wmma_f32_16x16x4_f32(false, a0, false, B[0][k4][0],
                                                 (short)0, c_lo, false, false);
    c_hi = __builtin_amdgcn_wmma_f32_16x16x4_f32(false, a0, false, B[0][k4][1],
                                                 (short)0, c_hi, false, false);
    c_lo = __builtin_amdgcn_wmma_f32_16x16x4_f32(false, a1, false, B[1][k4][0],
                                                 (short)0, c_lo, false, false);
    c_hi = __builtin_amdgcn_wmma_f32_16x16x4_f32(false, a1, false, B[1][k4][1],
                                                 (short)0, c_hi, false, false);
    c_lo = __builtin_amdgcn_wmma_f32_16x16x4_f32(false, a2, false, B[2][k4][0],
                                                 (short)0, c_lo, false, false);
    c_hi = __builtin_amdgcn_wmma_f32_16x16x4_f32(false, a2, false, B[2][k4][1],
                                                 (short)0, c_hi, false, false);
  }

  // Epilogue: C/D layout -> VGPR v, lanes 0-15: M=v, lanes 16-31: M=v+8; N=m
  const float b_lo = bias[m];
  const float b_hi = bias[16 + m];
#pragma unroll
  for (int v = 0; v < 8; ++v) {
    const int rown = tile * 16 + v + half * 8;
    float r_lo = c_lo[v] + b_lo;
    float r_hi = c_hi[v] + b_hi;
    if (resid) {
      r_lo += resid[(size_t)rown * DF + m];
      r_hi += resid[(size_t)rown * DF + 16 + m];
    }
    if (do_relu) {
      r_lo = fmaxf(r_lo, 0.0f);
      r_hi = fmaxf(r_hi, 0.0f);
    }
    out[(size_t)rown * DF + m]      = r_lo;
    out[(size_t)rown * DF + 16 + m] = r_hi;
  }
}

extern "C" void kernel_launch(void* const* d_in, const int* in_sizes, int n_in,
                              void* d_out, int out_size, void* d_ws, size_t ws_size,
                              hipStream_t stream) {
  const float* x  = (const float*)d_in[0];
  const int*   ei = (const int*)d_in[1];   // edge_index, 2 x E (harness int32)
  const float* W1 = (const float*)d_in[2];
  const float* b1 = (const float*)d_in[3];
  const float* W2 = (const float*)d_in[4];
  const float* b2 = (const float*)d_in[5];
  const int* row = ei;
  const int* col = ei + NE;

  float* ws   = (float*)d_ws;
  float* deg  = ws;                    // N
  float* dinv = deg  + NN;             // N
  float* w    = dinv + NN;             // E
  float* T1   = w    + NE;             // N*32
  float* P1   = T1   + (size_t)NN * DF;
  float* H    = P1   + (size_t)NN * DF;

  const int TPB = 256;
  const int eb  = (NE + TPB - 1) / TPB;
  const int nb  = (NN + TPB - 1) / TPB;
  const int pb  = (NE * 8 + TPB - 1) / TPB;
  const int tiles = (NN + 15) / 16;            // 6250 exactly
  const int gb  = (tiles + 7) / 8;             // 8 waves per 256-thr block

  // graph normalization (computed once, reused by all 4 prop passes)
  hipMemsetAsync(deg, 0, NN * sizeof(float), stream);
  deg_kernel<<<eb, TPB, 0, stream>>>(row, col, deg, NE);
  dinv_kernel<<<nb, TPB, 0, stream>>>(deg, dinv, NN);
  w_kernel<<<eb, TPB, 0, stream>>>(row, col, dinv, w, NE);

  // layer 1: h = relu(x@W1[0] + T1@W1[1] + (2*P1 - x)@W1[2] + b1)
  hipMemsetAsync(T1, 0, (size_t)NN * DF * sizeof(float), stream);
  prop_kernel<<<pb, TPB, 0, stream>>>(x, row, col, w, T1, NE);
  hipMemsetAsync(P1, 0, (size_t)NN * DF * sizeof(float), stream);
  prop_kernel<<<pb, TPB, 0, stream>>>(T1, row, col, w, P1, NE);
  cheb_wmma_kernel<<<gb, TPB, 0, stream>>>(x, T1, P1, W1, b1, nullptr, H, NN, 1);

  // layer 2: out = h@W2[0] + T1@W2[1] + (2*P1 - h)@W2[2] + b2 + x
  hipMemsetAsync(T1, 0, (size_t)NN * DF * sizeof(float), stream);
  prop_kernel<<<pb, TPB, 0, stream>>>(H, row, col, w, T1, NE);
  hipMemsetAsync(P1, 0, (size_t)NN * DF * sizeof(float), stream);
  prop_kernel<<<pb, TPB, 0, stream>>>(T1, row, col, w, P1, NE);
  cheb_wmma_kernel<<<gb, TPB, 0, stream>>>(H, T1, P1, W2, b2, x, (float*)d_out, NN, 0);
}